// NoiseMemoryBank_32512902431375
// MI455X (gfx1250) — compile-verified
//
#include <hip/hip_runtime.h>

typedef __attribute__((ext_vector_type(4))) float v4f;

// ---------------------------------------------------------------------------
// Hot kernel: bandwidth gather-copy for FEAT_DIM=256 (64 float4 chunks/row).
// One wave32 per 1 KiB row: lane l moves chunks {l, l+32}.
// Both `sample` (= g>>5) and the resolved `row` are wave-uniform by
// construction, so readfirstlane both: all base-pointer math is scalar
// (s_cselect_b64 between bank-row and fallback-row bases), the index fetch
// becomes a scalar load, and the payload is SADDR-form NT b128 with only a
// lane*16B vector offset. NT because the 512 MB bank > 192 MB L2 and every
// byte is touched at most once; output is write-once streaming.
// ---------------------------------------------------------------------------
__global__ void nmb_gather_rows64(const float* __restrict__ bank,
                                  const float* __restrict__ fallback,
                                  const int* __restrict__ rows,
                                  float* __restrict__ out,
                                  unsigned batch) {
  unsigned g = blockIdx.x * blockDim.x + threadIdx.x;
  // Wave-uniform sample id (32 lanes per row; blocks are wave-aligned).
  unsigned sample = __builtin_amdgcn_readfirstlane(g >> 5);
  if (sample >= batch) return;
  unsigned lane = threadIdx.x & 31u;

  // Wave-uniform resolved row (scalar address -> scalar load).
  int row = __builtin_amdgcn_readfirstlane(rows[sample]);

  // Fully scalar base selection: 256 floats (1 KiB) per row.
  const float* srcbase = (row >= 0)
      ? bank + ((size_t)(unsigned)row << 8)
      : fallback + ((size_t)sample << 8);
  float* dstbase = out + ((size_t)sample << 8);

  const v4f* __restrict__ src = (const v4f*)srcbase;
  v4f* __restrict__ dst = (v4f*)dstbase;

  v4f v0 = __builtin_nontemporal_load(src + lane);
  v4f v1 = __builtin_nontemporal_load(src + lane + 32);
  __builtin_nontemporal_store(v0, dst + lane);
  __builtin_nontemporal_store(v1, dst + lane + 32);
}

// ---------------------------------------------------------------------------
// Phase 1: resolve each sample's source row once into d_ws.
//   row = cid*CAPACITY + (rand_idx % cnt)   if cnt > 0
//   row = -1 (use fallback_noise)           if cnt == 0
// ---------------------------------------------------------------------------
__global__ void nmb_resolve_rows(const int* __restrict__ cids,
                                 const int* __restrict__ rand_idx,
                                 const int* __restrict__ counts,
                                 int* __restrict__ rows,
                                 int batch, int capacity) {
  int i = blockIdx.x * blockDim.x + threadIdx.x;
  if (i >= batch) return;
  int cid = cids[i];
  int cnt = counts[cid];
  int row = -1;
  if (cnt > 0) {
    unsigned r = (unsigned)rand_idx[i] % (unsigned)cnt;
    row = cid * capacity + (int)r;
  }
  rows[i] = row;
}

// Generic fallback for feature dims != 256 (not used with the given shapes).
__global__ void nmb_gather_rows_generic(const float* __restrict__ bank,
                                        const float* __restrict__ fallback,
                                        const int* __restrict__ rows,
                                        float* __restrict__ out,
                                        unsigned batch, unsigned chunks) {
  unsigned g = blockIdx.x * blockDim.x + threadIdx.x;
  unsigned sample = __builtin_amdgcn_readfirstlane(g >> 5);
  if (sample >= batch) return;
  unsigned lane = threadIdx.x & 31u;
  int row = __builtin_amdgcn_readfirstlane(rows[sample]);
  const v4f* __restrict__ src =
      (row >= 0) ? (const v4f*)bank + (size_t)(unsigned)row * chunks
                 : (const v4f*)fallback + (size_t)sample * chunks;
  v4f* __restrict__ dst = (v4f*)out + (size_t)sample * chunks;
  for (unsigned c = lane; c < chunks; c += 32) {
    v4f v = __builtin_nontemporal_load(src + c);
    __builtin_nontemporal_store(v, dst + c);
  }
}

extern "C" void kernel_launch(void* const* d_in, const int* in_sizes, int n_in,
                              void* d_out, int out_size, void* d_ws, size_t ws_size,
                              hipStream_t stream) {
  // setup_inputs() order:
  //   0: target_center_ids [B]        int32
  //   1: rand_idx          [B]        int32
  //   2: counts            [NC]       int32
  //   3: bank              [NC,CAP,D] fp32
  //   4: fallback_noise    [B,D]      fp32
  const int*   cids     = (const int*)d_in[0];
  const int*   rand_idx = (const int*)d_in[1];
  const int*   counts   = (const int*)d_in[2];
  const float* bank     = (const float*)d_in[3];
  const float* fallback = (const float*)d_in[4];
  float*       out      = (float*)d_out;

  const int batch       = in_sizes[0];
  const int num_centers = in_sizes[2];
  const int feat        = out_size / batch;                   // 256
  const int capacity    = in_sizes[3] / (num_centers * feat); // 512
  const int chunks      = feat / 4;                           // 64 float4 / row

  int* rows = (int*)d_ws;  // batch * 4 bytes of scratch

  {
    int threads = 256;
    int blocks  = (batch + threads - 1) / threads;
    nmb_resolve_rows<<<blocks, threads, 0, stream>>>(cids, rand_idx, counts,
                                                     rows, batch, capacity);
  }
  {
    unsigned total_threads = (unsigned)batch * 32u;  // one wave32 per row
    unsigned threads = 256;
    unsigned blocks  = (total_threads + threads - 1) / threads;
    if (chunks == 64) {
      nmb_gather_rows64<<<blocks, threads, 0, stream>>>(bank, fallback, rows,
                                                        out, (unsigned)batch);
    } else {
      nmb_gather_rows_generic<<<blocks, threads, 0, stream>>>(
          bank, fallback, rows, out, (unsigned)batch, (unsigned)chunks);
    }
  }
}